// WaymoPostProcessing_1683627180449
// MI455X (gfx1250) — compile-verified
//
#include <hip/hip_runtime.h>
#include <stdint.h>

// Waymo post-processing for MI455X (gfx1250).
//   S=8, A=64 agents -> 512 blocks of 64 threads (2 wave32 waves).
//   Per agent: TDM-gather xy stream -> LDS, transpose, pairwise dist^2 via
//   V_WMMA_F32_16X16X4_F32 rank-4 trick, hw v_sqrt + mean -> ADE,
//   wave-shuffle argmax NMS (MTR), serial 6x6 MPA, fused softmax.

#define S_DIM 8
#define A_DIM 64
#define P_DIM 32
#define T_DIM 80
#define K_PRED 6

typedef float v2f  __attribute__((ext_vector_type(2)));
typedef float v8f  __attribute__((ext_vector_type(8)));
typedef unsigned int u32x4 __attribute__((ext_vector_type(4)));
typedef int   i32x4 __attribute__((ext_vector_type(4)));
typedef int   i32x8 __attribute__((ext_vector_type(8)));

#if defined(__HIP_DEVICE_COMPILE__)
  #if __has_builtin(__builtin_amdgcn_wmma_f32_16x16x4_f32)
    #define USE_WMMA 1
  #else
    #define USE_WMMA 0
  #endif
  #if __has_builtin(__builtin_amdgcn_tensor_load_to_lds) && \
      __has_builtin(__builtin_amdgcn_s_wait_tensorcnt)
    #define USE_TDM 1
  #else
    #define USE_TDM 0
  #endif
#else
  #define USE_WMMA 0
  #define USE_TDM 0
#endif

// clamp-to-[0,inf) + hardware sqrt: v_med3_num_f32 + v_sqrt_f32 (2 VALU ops),
// avoiding the precise-libm sqrt expansion (~10 ops incl. scalar cmps).
__device__ __forceinline__ float sqrt_clamp_fast(float x) {
#if defined(__HIP_DEVICE_COMPILE__) && __has_builtin(__builtin_amdgcn_fmed3f) && \
    __has_builtin(__builtin_amdgcn_sqrtf)
  return __builtin_amdgcn_sqrtf(__builtin_amdgcn_fmed3f(x, 0.0f, 3.402823466e38f));
#else
  return sqrtf(fmaxf(x, 0.0f));
#endif
}

__global__ __launch_bounds__(64)
void waymo_post_kernel(const unsigned char* __restrict__ g_valid,   // (S,A) bool
                       const float* __restrict__ g_scores,          // (S,A,P)
                       const float* __restrict__ g_trajs,           // (S,A,P,T,4)
                       const float* __restrict__ g_atype,           // (S,A,3)
                       float* __restrict__ o_traj,                  // (S,T,A,K,2)
                       float* __restrict__ o_yaw,                   // (S,T,A,K,1)
                       float* __restrict__ o_spd,                   // (S,T,A,K,1)
                       float* __restrict__ o_sc,                    // (S,A,K)
                       float* __restrict__ o_val)                   // (S,T,A)
{
  const int sa   = blockIdx.x;
  const int s    = sa >> 6;          // / A_DIM
  const int a    = sa & 63;
  const int tid  = threadIdx.x;
  const int lane = tid & 31;
  const int wv   = tid >> 5;
  const int hh   = lane >> 4;        // half-wave selector (K-split / M-split)
  const int l15  = lane & 15;

  __shared__ __align__(16) float stage[P_DIM * T_DIM * 2]; // (p,t) xy pairs, 20KB
  __shared__ float2 xy[T_DIM][P_DIM];                      // transposed,     20KB
  __shared__ float  ade[P_DIM][P_DIM + 1];                 // padded rows
  __shared__ int    midx[K_PRED];
  __shared__ float  fin_lds[K_PRED];

  const float* src = g_trajs + (size_t)sa * (P_DIM * T_DIM * 4);

  // ---------------- Stage A: xy stream -> LDS (TDM async gather) -----------
#if USE_TDM
  if (wv == 0) {
    const uint64_t ga      = (uint64_t)(uintptr_t)src;
    const uint32_t lds_off = (uint32_t)(uintptr_t)(&stage[0]); // low32 = LDS byte offset
    u32x4 g0;
    g0.x = 0x1u;                                           // count=1, user mode
    g0.y = lds_off;                                        // lds_addr
    g0.z = (uint32_t)ga;                                   // global_addr[31:0]
    g0.w = ((uint32_t)(ga >> 32) & 0x01FFFFFFu) | 0x80000000u; // addr[56:32] | type=2
    i32x8 g1;
    g1[0] = 0x00030000;            // data_size = 3 (8-byte elements)
    g1[1] = 2 << 16;               // tensor_dim0 = 2  (xy , zw per record)
    g1[2] = (int)(6400u << 16);    // tensor_dim1 = 6400 rows
    g1[3] = 1 << 16;               // tile_dim0   = 1  (take xy only)
    g1[4] = 6400;                  // tile_dim1   = 6400, tile_dim2 = 0
    g1[5] = 2;                     // tensor_dim0_stride = 2 elements (16B)
    g1[6] = 0;
    g1[7] = 0;
    const i32x4 gz = {0, 0, 0, 0};
#if defined(__clang_major__) && (__clang_major__ >= 23)
    const i32x8 gz8 = {0, 0, 0, 0, 0, 0, 0, 0};
    __builtin_amdgcn_tensor_load_to_lds(g0, g1, gz, gz, gz8, 0);
#else
    __builtin_amdgcn_tensor_load_to_lds(g0, g1, gz, gz, 0);
#endif
    __builtin_amdgcn_s_wait_tensorcnt(0);
  }
#else
  {
    const float2* sxy = (const float2*)src;
    float2* st2 = (float2*)stage;
    for (int i = tid; i < P_DIM * T_DIM; i += 64)
      st2[i] = sxy[i * 2];   // xy of each (p,t) record
  }
#endif
  __syncthreads();

  // ---------------- Stage B: transpose to xy[t][p] -------------------------
  {
    const float2* st2 = (const float2*)stage;
    for (int i = tid; i < P_DIM * T_DIM; i += 64) {
      const int t = i >> 5;       // p fastest -> conflict-free xy writes
      const int p = i & 31;
      xy[t][p] = st2[p * T_DIM + t];
    }
  }
  __syncthreads();

  // ---------------- Stage C: pairwise ADE (WMMA rank-4 dist^2) -------------
#if USE_WMMA
  {
    v8f acc0 = {}; v8f acc1 = {};
    const int ti = wv;            // wave 0 -> rows 0..15, wave 1 -> rows 16..31
    for (int t = 0; t < T_DIM; ++t) {
      const float2 p0 = xy[t][l15];        // modes 0..15
      const float2 p1 = xy[t][16 + l15];   // modes 16..31
      const float2 am = ti ? p1 : p0;
      const float na = am.x * am.x + am.y * am.y;
      // A(16x4): row m = [x, y, 1, |p|^2]; lanes 0-15 carry K0/K1, 16-31 K2/K3
      v2f A;
      A.x = hh ? 1.0f : am.x;
      A.y = hh ? na   : am.y;
      const float n0 = p0.x * p0.x + p0.y * p0.y;
      const float n1 = p1.x * p1.x + p1.y * p1.y;
      // B(4x16): col n = [-2x, -2y, |p|^2, 1]^T
      v2f B0, B1;
      B0.x = hh ? n0   : (-2.0f * p0.x);
      B0.y = hh ? 1.0f : (-2.0f * p0.y);
      B1.x = hh ? n1   : (-2.0f * p1.x);
      B1.y = hh ? 1.0f : (-2.0f * p1.y);
      const v8f z = {};
      v8f d0 = __builtin_amdgcn_wmma_f32_16x16x4_f32(false, A, false, B0,
                                                     (short)0, z, false, false);
      v8f d1 = __builtin_amdgcn_wmma_f32_16x16x4_f32(false, A, false, B1,
                                                     (short)0, z, false, false);
#pragma unroll
      for (int e = 0; e < 8; ++e) {
        acc0[e] += sqrt_clamp_fast(d0[e]);
        acc1[e] += sqrt_clamp_fast(d1[e]);
      }
    }
#pragma unroll
    for (int e = 0; e < 8; ++e) {
      const int row = ti * 16 + e + 8 * hh;   // C/D layout: VGPR e, halves M/M+8
      ade[row][l15]      = acc0[e] * (1.0f / T_DIM);
      ade[row][16 + l15] = acc1[e] * (1.0f / T_DIM);
    }
  }
#else
  for (int pr = tid; pr < P_DIM * P_DIM; pr += 64) {
    const int i2 = pr >> 5, j2 = pr & 31;
    float acc = 0.0f;
    for (int t = 0; t < T_DIM; ++t) {
      const float2 pi = xy[t][i2], pj = xy[t][j2];
      const float dx = pi.x - pj.x, dy = pi.y - pj.y;
      acc += sqrt_clamp_fast(dx * dx + dy * dy);
    }
    ade[i2][j2] = acc * (1.0f / T_DIM);
  }
#endif
  __syncthreads();

  // ---------------- Stage D/E: MTR NMS + MPA NMS (wave 0) -------------------
  float thr_mtr, thr_mpa;
  {
    const float* at = g_atype + (size_t)sa * 3;
    const float a0 = at[0], a1 = at[1], a2 = at[2];
    thr_mtr = a0 * 3.5f + a1 * 1.0f + a2 * 1.75f;
    thr_mpa = a0 * 2.5f + a1 * 1.0f + a2 * 1.5f;
  }

  if (wv == 0) {
    // normalize scores over P (one mode per lane, P == wave32)
    float sraw = g_scores[(size_t)sa * P_DIM + lane];
    float ssum = sraw;
#pragma unroll
    for (int off = 16; off; off >>= 1) ssum += __shfl_xor(ssum, off, 32);
    const float scn = sraw / ssum;

    // MTR greedy NMS, K_PRED rounds of wave-argmax (first-index tie-break)
    float carry = scn;
    int my_mode = 0;
    for (int k = 0; k < K_PRED; ++k) {
      float bv = carry; int bi = lane;
#pragma unroll
      for (int off = 16; off; off >>= 1) {
        const float ov = __shfl_xor(bv, off, 32);
        const int   oi = __shfl_xor(bi, off, 32);
        if (ov > bv || (ov == bv && oi < bi)) { bv = ov; bi = oi; }
      }
      const int idx = bi;
      if (lane == k) my_mode = idx;
      if (lane == 0) midx[k] = idx;
      const bool win = ade[idx][lane] < thr_mtr;
      carry *= win ? 0.01f : 1.0f;
      if (lane == idx) carry = -1.0f;
    }

    // gather selected scores, renormalize over K
    const float skv = __shfl(scn, my_mode, 32);
    float s6 = (lane < K_PRED) ? skv : 0.0f;
#pragma unroll
    for (int off = 16; off; off >>= 1) s6 += __shfl_xor(s6, off, 32);
    const float sk_norm = skv / s6;

    float cur[K_PRED]; int mk[K_PRED];
#pragma unroll
    for (int k = 0; k < K_PRED; ++k) {
      cur[k] = __shfl(sk_norm, k, 32);
      mk[k]  = __shfl(my_mode, k, 32);
    }

    // stable descending order (== argsort(-s) with stable ties)
    int order[K_PRED]; bool used[K_PRED];
#pragma unroll
    for (int k = 0; k < K_PRED; ++k) used[k] = false;
    for (int i = 0; i < K_PRED; ++i) {
      float best = -1e30f; int bj = 0;
      for (int j = 0; j < K_PRED; ++j)
        if (!used[j] && cur[j] > best) { best = cur[j]; bj = j; }
      order[i] = bj; used[bj] = true;
    }

    const bool vld = g_valid[sa] != 0;
    for (int i = 0; i < K_PRED; ++i) {
      const int k = order[i];
      bool sup = false;
      for (int j = 0; j < K_PRED; ++j) {
        const bool w2 = ade[mk[k]][mk[j]] < thr_mpa;
        sup = sup || (w2 && (cur[j] > cur[k]));
      }
      if (vld && sup) cur[k] = 0.001f;
    }

    // softmax(log(s/sum)/0.8) == s^1.25 / sum(s^1.25)
    float tot = 0.0f, fin[K_PRED];
#pragma unroll
    for (int k = 0; k < K_PRED; ++k) {
      fin[k] = __expf(1.25f * __logf(cur[k]));
      tot += fin[k];
    }
    if (lane < K_PRED) fin_lds[lane] = fin[lane] / tot;
  }
  __syncthreads();

  // ---------------- Stage F: outputs ---------------------------------------
  const float vf = (g_valid[sa] != 0) ? 1.0f : 0.0f;
  const float4* src4 = (const float4*)src;
  for (int i = tid; i < T_DIM * K_PRED; i += 64) {
    const int t = i / K_PRED;
    const int k = i - t * K_PRED;
    const float4 e = src4[midx[k] * T_DIM + t];
    const size_t ob = (((size_t)s * T_DIM + t) * A_DIM + a) * K_PRED + k;
    o_traj[ob * 2 + 0] = e.x;
    o_traj[ob * 2 + 1] = e.y;
    o_yaw[ob] = e.z;
    o_spd[ob] = e.w;
  }
  if (tid < K_PRED) o_sc[(size_t)sa * K_PRED + tid] = fin_lds[tid];
  for (int t = tid; t < T_DIM; t += 64)
    o_val[((size_t)s * T_DIM + t) * A_DIM + a] = vf;
}

extern "C" void kernel_launch(void* const* d_in, const int* in_sizes, int n_in,
                              void* d_out, int out_size, void* d_ws, size_t ws_size,
                              hipStream_t stream) {
  (void)in_sizes; (void)n_in; (void)d_ws; (void)ws_size; (void)out_size;
  const unsigned char* g_valid  = (const unsigned char*)d_in[0]; // bool (S,A)
  const float*         g_scores = (const float*)d_in[1];         // (S,A,P)
  const float*         g_trajs  = (const float*)d_in[2];         // (S,A,P,T,4)
  const float*         g_atype  = (const float*)d_in[3];         // (S,A,3)

  float* out = (float*)d_out;
  float* o_traj = out;                                                 // S*T*A*K*2
  float* o_yaw  = o_traj + (size_t)S_DIM * T_DIM * A_DIM * K_PRED * 2; // S*T*A*K
  float* o_spd  = o_yaw  + (size_t)S_DIM * T_DIM * A_DIM * K_PRED;    // S*T*A*K
  float* o_sc   = o_spd  + (size_t)S_DIM * T_DIM * A_DIM * K_PRED;    // S*A*K
  float* o_val  = o_sc   + (size_t)S_DIM * A_DIM * K_PRED;            // S*T*A

  waymo_post_kernel<<<dim3(S_DIM * A_DIM), dim3(64), 0, stream>>>(
      g_valid, g_scores, g_trajs, g_atype, o_traj, o_yaw, o_spd, o_sc, o_val);
}